// Encoder_1348619731390
// MI455X (gfx1250) — compile-verified
//
#include <hip/hip_runtime.h>
#include <hip/hip_bf16.h>
#include <math.h>
#include <stdint.h>

// Problem constants (from the reference)
constexpr int kN = 2000;
constexpr int kC = 8;
constexpr int kV = 6;
constexpr int kD = 128;
constexpr int kG = 1000;
constexpr int kL = 5;
constexpr int kK = 6;

typedef __attribute__((ext_vector_type(2))) float v2f;
typedef __attribute__((ext_vector_type(8))) float v8f;

#define LDS_STRIDE 132   // 128 + 4 pad to avoid 16-way LDS bank conflicts

// ---------------------------------------------------------------------------
// Gather-GEMM with fused combine epilogue.
//   out[r, 0:128] over K = nBlk*128, where A's K-block `kb` for row r is the
//   128-float row  table[srcIdx[r*nBlk+kb]]  (srcIdx==null -> identity row r).
// MODE 0:  out = x@W1 + b1                          (negation transform)
// MODE 1:  out = normalize(sigmoid(x@W1+b1) + x@W2 + b2)   (_combine)
//
// Block: 256 threads (8 waves); each wave computes one 16x16 output tile
// (two accumulators for W1/W2) over the full K dimension using
// V_WMMA_F32_16X16X4_F32.  The gathered A tile is staged into LDS with the
// CDNA5 async global->LDS copy (ASYNCcnt) and double-buffered so the copy of
// block kb+1 overlaps the WMMA work on block kb.
// ---------------------------------------------------------------------------
template<int MODE>
__global__ __launch_bounds__(256)
void enc_gemm_gather(const float* __restrict__ table,
                     const int* __restrict__ srcIdx,
                     const float* __restrict__ W1, const float* __restrict__ b1,
                     const float* __restrict__ W2, const float* __restrict__ b2,
                     float* __restrict__ out, int M, int nBlk)
{
    __shared__ float ldsA[2][16 * LDS_STRIDE];
    __shared__ float ldsO[16 * LDS_STRIDE];
    __shared__ float ldsInv[16];

    const int tid  = threadIdx.x;
    const int wave = tid >> 5;
    const int lane = tid & 31;
    const int m0   = blockIdx.x * 16;
    const int col0 = wave * 16;
    const int half = (lane < 16) ? 0 : 2;   // second lane-half holds K+2
    const int nl   = lane & 15;

    // Per-thread A staging slot: row r (0..15), 8 floats starting at d0.
    const int rA  = tid >> 4;
    const int d0A = (tid & 15) * 8;
    int rowA = m0 + rA;
    if (rowA >= M) rowA = M - 1;           // clamp; clamped rows never stored
    const uint32_t dstBase[2] = {
        (uint32_t)(uintptr_t)&ldsA[0][rA * LDS_STRIDE + d0A],
        (uint32_t)(uintptr_t)&ldsA[1][rA * LDS_STRIDE + d0A]
    };

    // Async gather of one 16x128 A block into LDS buffer `buf`.
    auto issueA = [&](int kb, int buf) {
        const int src = srcIdx ? srcIdx[(size_t)rowA * nBlk + kb] : rowA;
        const float* g = table + (size_t)src * kD + d0A;
        const uint32_t dst = dstBase[buf];
        asm volatile("global_load_async_to_lds_b128 %0, %1, off"
                     :: "v"(dst), "v"(g) : "memory");
        asm volatile("global_load_async_to_lds_b128 %0, %1, off"
                     :: "v"(dst + 16u), "v"(g + 4) : "memory");
    };

    v8f c1 = {};
    v8f c2 = {};

    issueA(0, 0);
    for (int kb = 0; kb < nBlk; ++kb) {
        const int cur = kb & 1;
        if (kb + 1 < nBlk) {
            issueA(kb + 1, cur ^ 1);
            // our 2 newest async loads may stay outstanding; block kb's done
            asm volatile("s_wait_asynccnt 0x2" ::: "memory");
        } else {
            asm volatile("s_wait_asynccnt 0x0" ::: "memory");
        }
        __syncthreads();                    // all waves' tile kb visible

        const float* W1b = W1 + (size_t)kb * kD * kD;
        const float* W2b = MODE ? (W2 + (size_t)kb * kD * kD) : nullptr;
        const float* ldsArow = &ldsA[cur][nl * LDS_STRIDE];

        // ---- 32 WMMA steps of K=4 over this 128-wide block ----
#pragma unroll 4
        for (int kk = 0; kk < 32; ++kk) {
            const int k0 = kk * 4;
            // A fragment: 16x4 f32. VGPR0: K=k0 (lanes<16) / K=k0+2 ; VGPR1: +1
            v2f a;
            a.x = ldsArow[k0 + half];
            a.y = ldsArow[k0 + half + 1];
            // B fragment: 4x16 f32. VGPR j: K = k0 + half + j, N = col0+nl
            const int kg = k0 + half;
            v2f bA;
            bA.x = W1b[(size_t)kg       * kD + col0 + nl];
            bA.y = W1b[(size_t)(kg + 1) * kD + col0 + nl];
            c1 = __builtin_amdgcn_wmma_f32_16x16x4_f32(
                     false, a, false, bA, (short)0, c1, false, false);
            if (MODE) {
                v2f bB;
                bB.x = W2b[(size_t)kg       * kD + col0 + nl];
                bB.y = W2b[(size_t)(kg + 1) * kD + col0 + nl];
                c2 = __builtin_amdgcn_wmma_f32_16x16x4_f32(
                         false, a, false, bB, (short)0, c2, false, false);
            }
        }
        __syncthreads();   // everyone done reading ldsA[cur] before reuse
    }

    // ---- epilogue: bias (+ sigmoid-add for MODE 1), stage to LDS ----
    {
        const int rbase = (lane < 16) ? 0 : 8;   // C/D: VGPR i -> row i / i+8
        const int coln  = col0 + nl;
        const float bb1 = b1[coln];
        const float bb2 = MODE ? b2[coln] : 0.f;
#pragma unroll
        for (int i = 0; i < 8; ++i) {
            const int r = rbase + i;
            float h1 = c1[i] + bb1;
            float val;
            if (MODE) {
                float h2 = c2[i] + bb2;
                val = 1.f / (1.f + __expf(-h1)) + h2;
            } else {
                val = h1;
            }
            ldsO[r * LDS_STRIDE + coln] = val;
        }
    }
    __syncthreads();

    if (MODE) {
        // per-row L2 normalize: x / max(||x||, 1e-12)
        if (tid < 16) {
            float ss = 0.f;
            for (int d = 0; d < kD; ++d) {
                float x = ldsO[tid * LDS_STRIDE + d];
                ss += x * x;
            }
            ldsInv[tid] = 1.f / fmaxf(sqrtf(ss), 1e-12f);
        }
        __syncthreads();
        const int row = m0 + rA;
        if (row < M) {
            const float inv = ldsInv[rA];
            float* o = out + (size_t)row * kD + d0A;
#pragma unroll
            for (int q = 0; q < 8; ++q)
                o[q] = ldsO[rA * LDS_STRIDE + d0A + q] * inv;
        }
    } else {
        const int row = m0 + rA;
        if (row < M) {
            float* o = out + (size_t)row * kD + d0A;
#pragma unroll
            for (int q = 0; q < 8; ++q)
                o[q] = ldsO[rA * LDS_STRIDE + d0A + q];
        }
    }
}

// ---------------------------------------------------------------------------
// Initial variables: rows 0..G-1 = clipped-norm embeddings, rows G..N-1 =
// normalized tseitin vector.
// ---------------------------------------------------------------------------
__global__ void enc_init_vars(const float* __restrict__ emb,
                              const float* __restrict__ tsei,
                              float* __restrict__ Tbuf)
{
    const int n = blockIdx.x;
    const int d = threadIdx.x;       // 128 threads
    __shared__ float red[kD];
    float x = (n < kG) ? emb[(size_t)n * kD + d] : tsei[d];
    red[d] = x * x;
    __syncthreads();
    for (int s = 64; s > 0; s >>= 1) {
        if (d < s) red[d] += red[d + s];
        __syncthreads();
    }
    const float nrm   = sqrtf(red[0]);
    const float scale = (n < kG) ? fminf(1.f, 1.f / (nrm + 1e-7f))
                                 : 1.f / fmaxf(nrm, 1e-12f);
    Tbuf[(size_t)n * kD + d] = x * scale;
}

// true_vec = false_vec @ Wn + bn   (128 threads, 1 block)
__global__ void enc_true_vec(const float* __restrict__ fv,
                             const float* __restrict__ Wn,
                             const float* __restrict__ bn,
                             float* __restrict__ tv)
{
    const int d = threadIdx.x;
    float s = bn[d];
    for (int k = 0; k < kD; ++k) s += fv[k] * Wn[k * kD + d];
    tv[d] = s;
}

__global__ void enc_copy_row(float* __restrict__ dst, const float* __restrict__ src)
{
    dst[threadIdx.x] = src[threadIdx.x];
}

// ---------------------------------------------------------------------------
// Source-row index for the variable-stage gather GEMM.
// Row r = ((n*K + k)*2B + j), block v.  order_org / order_perm + negation
// select + false padding are folded into a flat row id over the table
// [variables (N*B) | negated variables (N*B) | false_vec].
// ---------------------------------------------------------------------------
__global__ void enc_build_idx_v(const int* __restrict__ var_idx,
                                const unsigned char* __restrict__ neg_mask,
                                const int* __restrict__ pos,
                                int* __restrict__ idxV, int B, int total)
{
    int e = blockIdx.x * 256 + threadIdx.x;
    if (e >= total) return;
    const int v = e % kV;
    int t = e / kV;
    const int twoB = 2 * B;
    const int j = t % twoB; t /= twoB;
    const int k = t % kK;
    const int n = t / kK;

    const int p = pos[n * kC + k];          // position in [0, L)
    int s;
    if (v == 0) {
        s = p;                              // both orders start with p
    } else {
        // others = ascending indices != p;  org: others[v-1], perm: others[v%(V-1)]
        const int i = (j < B) ? (v - 1) : (v % (kV - 1));
        s = (i < p) ? i : i + 1;
    }
    const int b = (j < B) ? j : j - B;

    int src;
    if (s < kL) {
        const int base = (n * kC + k) * kV + s;
        const int vidx = var_idx[base];
        const int ng   = neg_mask[base];
        src = (ng ? kN * B : 0) + vidx * B + b;
    } else {
        src = 2 * kN * B;                   // false_vec row
    }
    idxV[e] = src;
}

// ---------------------------------------------------------------------------
// Source-row index for the clause-stage gather GEMM.
// Row r = (n*4B + j), block c.  Rows j<2B read cl, rows j>=2B read
// roll(cl,-1) (clause c -> (c+1)%C).  Clauses >= K read the true_vec row.
// ---------------------------------------------------------------------------
__global__ void enc_build_idx_c(int* __restrict__ idxC, int B, int total, int trueRow)
{
    int e = blockIdx.x * 256 + threadIdx.x;
    if (e >= total) return;
    const int c = e % kC;
    int t = e / kC;
    const int fourB = 4 * B;
    const int j = t % fourB;
    const int n = t / fourB;
    const int twoB = 2 * B;
    const int cp = (j < twoB) ? c : (c + 1) % kC;
    const int jj = (j < twoB) ? j : j - twoB;
    idxC[e] = (cp < kK) ? ((n * kK + cp) * twoB + jj) : trueRow;
}

// ---------------------------------------------------------------------------
// Final: copy variables (N,16,128) and compute
// aux[n] = sum_b || variables[n,b,:] - mean_b ||_2
// ---------------------------------------------------------------------------
__global__ void enc_finalize(const float* __restrict__ Tbuf,
                             float* __restrict__ out_vars,
                             float* __restrict__ out_aux)
{
    const int n = blockIdx.x;
    const int d = threadIdx.x;    // 128 threads
    __shared__ float red[kD];
    __shared__ float auxAcc;
    const float* vb = Tbuf + (size_t)n * 16 * kD;

    float vals[16];
    float mean = 0.f;
#pragma unroll
    for (int b = 0; b < 16; ++b) {
        vals[b] = vb[(size_t)b * kD + d];
        mean += vals[b];
    }
    mean *= (1.f / 16.f);
    if (d == 0) auxAcc = 0.f;
    __syncthreads();

    for (int b = 0; b < 16; ++b) {
        const float x = vals[b] - mean;
        red[d] = x * x;
        __syncthreads();
        for (int s = 64; s > 0; s >>= 1) {
            if (d < s) red[d] += red[d + s];
            __syncthreads();
        }
        if (d == 0) auxAcc += sqrtf(red[0]);
        __syncthreads();
        out_vars[((size_t)n * 16 + b) * kD + d] = vals[b];
    }
    if (d == 0) out_aux[n] = auxAcc;
}

// ---------------------------------------------------------------------------
extern "C" void kernel_launch(void* const* d_in, const int* in_sizes, int n_in,
                              void* d_out, int out_size, void* d_ws, size_t ws_size,
                              hipStream_t stream)
{
    const int*           var_idx  = (const int*)d_in[0];
    const unsigned char* neg_mask = (const unsigned char*)d_in[1];  // bool -> 1B
    const int*           pos      = (const int*)d_in[2];
    const float*         emb      = (const float*)d_in[3];
    const float*         tsei     = (const float*)d_in[4];
    const float*         fv       = (const float*)d_in[5];
    const float*         Wn       = (const float*)d_in[6];
    const float*         bn       = (const float*)d_in[7];
    const float*         W1v      = (const float*)d_in[8];
    const float*         b1v      = (const float*)d_in[9];
    const float*         W2v      = (const float*)d_in[10];
    const float*         b2v      = (const float*)d_in[11];
    const float*         W1c      = (const float*)d_in[12];
    const float*         b1c      = (const float*)d_in[13];
    const float*         W2c      = (const float*)d_in[14];
    const float*         b2c      = (const float*)d_in[15];

    // Workspace layout
    char* ws = (char*)d_ws;
    size_t off = 0;
    float* Tbuf = (float*)(ws + off);                  // variables|varNeg|false
    off += (size_t)(2 * kN * 16 + 2) * kD * sizeof(float);
    float* CE   = (float*)(ws + off);                  // clause embeddings + true row
    off += (size_t)(kN * kK * 8 + 1) * kD * sizeof(float);
    float* tv   = (float*)(ws + off);                  // true_vec
    off += (size_t)kD * sizeof(float);
    int*   idxV = (int*)(ws + off);
    off += (size_t)kN * kK * 8 * kV * sizeof(int);
    int*   idxC = (int*)(ws + off);
    off += (size_t)kN * 16 * kC * sizeof(int);

    enc_init_vars<<<kN, kD, 0, stream>>>(emb, tsei, Tbuf);
    enc_true_vec<<<1, kD, 0, stream>>>(fv, Wn, bn, tv);

    int B = 1;
    for (int t = 0; t < 2; ++t) {
        const int Mv = kN * B;
        // negated variables: varNeg = variables @ Wn + bn  (identity gather)
        enc_gemm_gather<0><<<(Mv + 15) / 16, 256, 0, stream>>>(
            Tbuf, nullptr, Wn, bn, nullptr, nullptr,
            Tbuf + (size_t)Mv * kD, Mv, 1);
        // false_vec row appended after varNeg region
        enc_copy_row<<<1, kD, 0, stream>>>(Tbuf + (size_t)2 * Mv * kD, fv);

        // variable -> clause combine
        const int totV = kN * kK * 2 * B * kV;
        enc_build_idx_v<<<(totV + 255) / 256, 256, 0, stream>>>(
            var_idx, neg_mask, pos, idxV, B, totV);
        const int M1 = kN * kK * 2 * B;
        enc_gemm_gather<1><<<(M1 + 15) / 16, 256, 0, stream>>>(
            Tbuf, idxV, W1v, b1v, W2v, b2v, CE, M1, kV);
        enc_copy_row<<<1, kD, 0, stream>>>(CE + (size_t)M1 * kD, tv);

        // clause -> variable combine
        const int totC = kN * 4 * B * kC;
        enc_build_idx_c<<<(totC + 255) / 256, 256, 0, stream>>>(idxC, B, totC, M1);
        const int M2 = kN * 4 * B;
        enc_gemm_gather<1><<<(M2 + 15) / 16, 256, 0, stream>>>(
            CE, idxC, W1c, b1c, W2c, b2c, Tbuf, M2, kC);

        B *= 4;
    }

    float* out_vars = (float*)d_out;
    float* out_aux  = out_vars + (size_t)kN * 16 * kD;
    enc_finalize<<<kN, kD, 0, stream>>>(Tbuf, out_vars, out_aux);
}